// Isomorphic_Feature_Extraction_5196910428808
// MI455X (gfx1250) — compile-verified
//
#include <hip/hip_runtime.h>
#include <hip/hip_bf16.h>
#include <math.h>

typedef __attribute__((ext_vector_type(2))) float v2f;
typedef __attribute__((ext_vector_type(8))) float v8f;

#define B_ 8
#define S_ 3600          // 60*60 spatial positions
#define CP_ 2            // c_prev
#define LC_ 8            // number of kernels
#define FAC_ 120         // 5!
#define NM_ 960          // LC_*FAC_
#define KPAD_ 28         // 25 padded to multiple of 4
#define NROWS_ 57600     // B_*S_*CP_
#define LDS_STRIDE_ 972  // 960 + 12 padding (bank-conflict break)

// ---------------------------------------------------------------------------
// Build M[m][q] = kernel[l][p_f(a)*5 + p_f(d)]  (m = l*120+f, q = a*5+d),
// padded to 28 cols, plus Msq[m] = sum(M^2).
// Permutation decoded from index f in factorial base => lexicographic order,
// matching itertools.permutations(range(5)).
// ---------------------------------------------------------------------------
__global__ void mprep_kernel(const float* __restrict__ kern,
                             float* __restrict__ Mmat,
                             float* __restrict__ Msq) {
    int m = blockIdx.x * blockDim.x + threadIdx.x;
    if (m >= NM_) return;
    int l = m / FAC_;
    int f = m % FAC_;
    int avail[5] = {0, 1, 2, 3, 4};
    int p[5];
    const int fact[5] = {24, 6, 2, 1, 1};
    int idx = f;
    #pragma unroll
    for (int t = 0; t < 5; ++t) {
        int d = idx / fact[t];
        idx %= fact[t];
        p[t] = avail[d];
        for (int u = d; u < 4 - t; ++u) avail[u] = avail[u + 1];
    }
    const float* kl = kern + l * 25;
    float* mr = Mmat + (size_t)m * KPAD_;
    float acc = 0.0f;
    #pragma unroll
    for (int a = 0; a < 5; ++a)
        #pragma unroll
        for (int d = 0; d < 5; ++d) {
            float v = kl[p[a] * 5 + p[d]];
            mr[a * 5 + d] = v;
            acc += v * v;
        }
    mr[25] = 0.0f; mr[26] = 0.0f; mr[27] = 0.0f;
    Msq[m] = acc;
}

// ---------------------------------------------------------------------------
// Materialize patch rows pf[row][28] (row = (b*S + s)*2 + cp) and xsq[row].
// ---------------------------------------------------------------------------
__global__ void patches_kernel(const float* __restrict__ x,
                               float* __restrict__ pf,
                               float* __restrict__ xsq) {
    int row = blockIdx.x * blockDim.x + threadIdx.x;
    if (row >= NROWS_) return;
    int cp = row & 1;
    int bs = row >> 1;
    int b = bs / S_;
    int s = bs % S_;
    int i = s / 60, j = s % 60;
    const float* xp = x + ((((size_t)b * CP_ + cp) * 64) + i) * 64 + j;
    float* pr = pf + (size_t)row * KPAD_;
    float acc = 0.0f;
    #pragma unroll
    for (int a = 0; a < 5; ++a)
        #pragma unroll
        for (int d = 0; d < 5; ++d) {
            float v = xp[a * 64 + d];
            pr[a * 5 + d] = v;
            acc += v * v;
        }
    pr[25] = 0.0f; pr[26] = 0.0f; pr[27] = 0.0f;
    xsq[row] = acc;
}

// ---------------------------------------------------------------------------
// Fused GEMM (f32 WMMA, 16x16x4, K=28) + dist + min over f per l.
// One workgroup = one 16-row tile x all 960 cols; 4 waves x 15 col-tiles.
// dist staged in LDS (16 x 972 floats ~= 62 KB), then 128 threads each
// min-reduce one (row, l) pair over its 120 contiguous f values.
// feature written directly in (b, c, s) layout for the softmax stage.
// ---------------------------------------------------------------------------
__global__ __launch_bounds__(128) void gemm_min_kernel(
        const float* __restrict__ pf, const float* __restrict__ xsq,
        const float* __restrict__ Mmat, const float* __restrict__ Msq,
        float* __restrict__ feature) {
    __shared__ float dist_lds[16 * LDS_STRIDE_];

    const int tile  = blockIdx.x;        // 0..3599
    const int r0    = tile * 16;
    const int lane  = threadIdx.x & 31;
    const int wave  = threadIdx.x >> 5;  // 0..3
    const int lhalf = lane >> 4;         // 0 or 1
    const int lmod  = lane & 15;

    // A operand: 7 K-chunks, float2 per lane.
    // Lanes 0-15: rows r0+lane, K = {4kk, 4kk+1}; lanes 16-31: K = {4kk+2, 4kk+3}.
    v2f a[7];
    const float* aptr = pf + (size_t)(r0 + lmod) * KPAD_ + 2 * lhalf;
    #pragma unroll
    for (int kk = 0; kk < 7; ++kk)
        a[kk] = *(const v2f*)(aptr + kk * 4);

    const float xsq_lo = xsq[r0 + lhalf * 8 + 0];  // unused; real loads below

    for (int nt = wave; nt < 60; nt += 4) {
        const int n0 = nt * 16;
        v8f c = {};
        const float* bptr = Mmat + (size_t)(n0 + lmod) * KPAD_ + 2 * lhalf;
        #pragma unroll
        for (int kk = 0; kk < 7; ++kk) {
            v2f b = *(const v2f*)(bptr + kk * 4);
            c = __builtin_amdgcn_wmma_f32_16x16x4_f32(
                    /*neg_a=*/false, a[kk], /*neg_b=*/false, b,
                    /*c_mod=*/(short)0, c, /*reuse_a=*/false, /*reuse_b=*/false);
        }
        const float msq = Msq[n0 + lmod];
        #pragma unroll
        for (int v = 0; v < 8; ++v) {
            const int rit = v + 8 * lhalf;                 // row within tile
            const float d = msq + xsq[r0 + rit] - 2.0f * c[v];
            dist_lds[rit * LDS_STRIDE_ + n0 + lmod] = d;
        }
    }
    (void)xsq_lo;
    __syncthreads();

    // Min reduction: thread t -> (row = t>>3, l = t&7), scan 120 f's.
    const int row = threadIdx.x >> 3;
    const int l   = threadIdx.x & 7;
    const float* dl = &dist_lds[row * LDS_STRIDE_ + l * FAC_];
    float mn = dl[0];
    #pragma unroll 8
    for (int f = 1; f < FAC_; ++f) mn = fminf(mn, dl[f]);

    const int grow = r0 + row;           // (b*S + s)*2 + cp
    const int cp   = grow & 1;
    const int bs   = grow >> 1;
    const int b    = bs / S_;
    const int s    = bs % S_;
    const int cidx = cp * LC_ + l;       // channel 0..15
    feature[((size_t)(b * 16 + cidx)) * S_ + s] = mn;
}

// ---------------------------------------------------------------------------
// Softmax over s (3600) per (b, c): out = exp(-f - max(-f)) / sum.
// ---------------------------------------------------------------------------
__global__ __launch_bounds__(256) void softmax_kernel(
        const float* __restrict__ feature, float* __restrict__ out) {
    const int bc = blockIdx.x;  // 0..127
    const float* f = feature + (size_t)bc * S_;
    float* o = out + (size_t)bc * S_;
    __shared__ float red[256];

    float lmax = -3.402823466e38f;
    for (int s = threadIdx.x; s < S_; s += 256) lmax = fmaxf(lmax, -f[s]);
    red[threadIdx.x] = lmax;
    __syncthreads();
    for (int off = 128; off > 0; off >>= 1) {
        if (threadIdx.x < off)
            red[threadIdx.x] = fmaxf(red[threadIdx.x], red[threadIdx.x + off]);
        __syncthreads();
    }
    const float gmax = red[0];
    __syncthreads();

    float lsum = 0.0f;
    for (int s = threadIdx.x; s < S_; s += 256) lsum += expf(-f[s] - gmax);
    red[threadIdx.x] = lsum;
    __syncthreads();
    for (int off = 128; off > 0; off >>= 1) {
        if (threadIdx.x < off)
            red[threadIdx.x] += red[threadIdx.x + off];
        __syncthreads();
    }
    const float inv = 1.0f / red[0];

    for (int s = threadIdx.x; s < S_; s += 256)
        o[s] = expf(-f[s] - gmax) * inv;
}

// Passthrough: second tuple element (kernel, 200 floats) at tail of d_out.
__global__ void copytail_kernel(const float* __restrict__ kern,
                                float* __restrict__ out) {
    int i = threadIdx.x;
    if (i < LC_ * 25) out[(size_t)B_ * 16 * S_ + i] = kern[i];
}

extern "C" void kernel_launch(void* const* d_in, const int* in_sizes, int n_in,
                              void* d_out, int out_size, void* d_ws, size_t ws_size,
                              hipStream_t stream) {
    const float* x    = (const float*)d_in[0];  // (8,2,64,64)
    const float* kern = (const float*)d_in[1];  // (8,5,5)
    float* out = (float*)d_out;                 // 460800 layer + 200 kernel
    float* ws  = (float*)d_ws;

    // Workspace layout (floats), total ~8.64 MB:
    float* Mmat = ws;                        // 960*28   = 26880
    float* Msq  = ws + 26880;                // 960      -> 27840
    float* pf   = ws + 27840;                // 57600*28 = 1612800 -> 1640640
    float* xsq  = ws + 1640640;              // 57600    -> 1698240
    float* feat = ws + 1698240;              // 460800   -> 2159040

    mprep_kernel<<<(NM_ + 127) / 128, 128, 0, stream>>>(kern, Mmat, Msq);
    patches_kernel<<<(NROWS_ + 255) / 256, 256, 0, stream>>>(x, pf, xsq);
    gemm_min_kernel<<<NROWS_ / 16, 128, 0, stream>>>(pf, xsq, Mmat, Msq, feat);
    softmax_kernel<<<B_ * 16, 256, 0, stream>>>(feat, out);
    copytail_kernel<<<1, 256, 0, stream>>>(kern, out);
}